// DGCNN_26706106646594
// MI455X (gfx1250) — compile-verified
//
#include <hip/hip_runtime.h>

// ============================================================================
// DGCNN forward for MI455X (gfx1250, wave32, WMMA f32 16x16x4).
//
// Edge-conv restructured via linearity:
//   y[o,n,k] = A[o, idx[n,k]] + Bv[o,n],  A = w1·X, Bv = (w2-w1)·X
// All heavy math is dense f32 GEMMs on v_wmma_f32_16x16x4_f32, register-
// blocked 16x64 per wave (A fragment reused across 4 B fragments; 4
// independent accumulators so back-to-back WMMAs have no RAW hazard).
// KNN distances use the same WMMA path (S = X^T X) fused with the
// 2S - xx - xx^T epilogue; fp32 throughout to preserve neighbor ordering.
// ============================================================================

typedef __attribute__((ext_vector_type(2))) float v2f;
typedef __attribute__((ext_vector_type(8))) float v8f;

#define BATCH 8
#define NPTS  1024
#define KNN   20
#define NEG_INF (-3.4e38f)

__device__ __forceinline__ float lrelu(float v) { return v >= 0.f ? v : 0.01f * v; }

// ---- x (B,N,3) -> P (B,4,N), channel 3 zero-padded (so K is a multiple of 4)
__global__ void k_pad_transpose(const float* __restrict__ x, float* __restrict__ P) {
    int t = blockIdx.x * blockDim.x + threadIdx.x;
    if (t >= BATCH * 4 * NPTS) return;
    int n = t % NPTS;
    int c = (t / NPTS) & 3;
    int b = t / (4 * NPTS);
    P[t] = (c < 3) ? x[((size_t)b * NPTS + n) * 3 + c] : 0.f;
}

// ---- xx[b,n] = sum_c X[b, ch_off+c, n]^2
__global__ void k_sqnorm(const float* __restrict__ X, int ch_off, int row_stride,
                         int batch_stride, int C, float* __restrict__ xx) {
    int t = blockIdx.x * blockDim.x + threadIdx.x;
    if (t >= BATCH * NPTS) return;
    int n = t % NPTS;
    int b = t / NPTS;
    const float* p = X + (size_t)b * batch_stride + (size_t)ch_off * row_stride + n;
    float s = 0.f;
    for (int c = 0; c < C; ++c) {
        float v = p[(size_t)c * row_stride];
        s += v * v;
    }
    xx[t] = s;
}

// ---- dist[b,m,n] = 2*sum_c X[c,m]X[c,n] - xx[m] - xx[n]
// WMMA f32 16x16x4, 16(m) x 64(n) tile per wave, 4 independent accumulators.
// grid: (N/64, N/16, B), block: 32 (EXEC all ones at every WMMA)
__global__ void k_dist_wmma(const float* __restrict__ X, int ch_off, int row_stride,
                            int batch_stride, int K4, const float* __restrict__ xx,
                            float* __restrict__ dist) {
    int n0 = blockIdx.x * 64;
    int m0 = blockIdx.y * 16;
    int b  = blockIdx.z;
    int lane = threadIdx.x;
    int r  = lane & 15;
    int hi = lane >> 4;
    const float* Xb = X + (size_t)b * batch_stride + (size_t)ch_off * row_stride;
    v8f acc[4] = {};
    for (int kc = 0; kc < K4; kc += 4) {
        int k0 = kc + 2 * hi;
        const float* r0 = Xb + (size_t)k0 * row_stride;
        const float* r1 = r0 + row_stride;
        v2f a;
        a.x = r0[m0 + r];
        a.y = r1[m0 + r];
#pragma unroll
        for (int j = 0; j < 4; ++j) {
            v2f bb;
            bb.x = r0[n0 + 16 * j + r];
            bb.y = r1[n0 + 16 * j + r];
            acc[j] = __builtin_amdgcn_wmma_f32_16x16x4_f32(false, a, false, bb,
                                                           (short)0, acc[j], false, false);
        }
    }
    float xm[8];
#pragma unroll
    for (int v = 0; v < 8; ++v) xm[v] = xx[b * NPTS + m0 + v + 8 * hi];
#pragma unroll
    for (int j = 0; j < 4; ++j) {
        float xn = xx[b * NPTS + n0 + 16 * j + r];
#pragma unroll
        for (int v = 0; v < 8; ++v) {
            int m = v + 8 * hi;
            float d = 2.f * acc[j][v] - xm[v] - xn;
            dist[((size_t)b * NPTS + (m0 + m)) * NPTS + n0 + 16 * j + r] = d;
        }
    }
}

// ---- top-k(20) indices of each dist row, register insertion sort, f4 loads
__global__ void k_topk(const float* __restrict__ dist, int* __restrict__ idx) {
    int t = blockIdx.x * blockDim.x + threadIdx.x;
    if (t >= BATCH * NPTS) return;
    const float4* row4 = (const float4*)(dist + (size_t)t * NPTS);
    float vv[KNN];
    int   vi[KNN];
#pragma unroll
    for (int k = 0; k < KNN; ++k) { vv[k] = NEG_INF; vi[k] = 0; }
    for (int j4 = 0; j4 < NPTS / 4; ++j4) {
        float4 d4 = row4[j4];
        float dv[4] = { d4.x, d4.y, d4.z, d4.w };
#pragma unroll
        for (int e = 0; e < 4; ++e) {
            float d = dv[e];
            if (d > vv[KNN - 1]) {
                vv[KNN - 1] = d; vi[KNN - 1] = 4 * j4 + e;
#pragma unroll
                for (int k = KNN - 1; k > 0; --k) {
                    if (vv[k] > vv[k - 1]) {
                        float tv = vv[k]; vv[k] = vv[k - 1]; vv[k - 1] = tv;
                        int   ti = vi[k]; vi[k] = vi[k - 1]; vi[k - 1] = ti;
                    }
                }
            }
        }
    }
#pragma unroll
    for (int k = 0; k < KNN; ++k) idx[(size_t)t * KNN + k] = vi[k];
}

// ---- stacked weights: Wp[0:O) = w1,  Wp[O:2O) = w2 - w1, K zero-padded
__global__ void k_prep_w(const float* __restrict__ w, int O, int C, int Kpad,
                         float* __restrict__ Wp) {
    int t = blockIdx.x * blockDim.x + threadIdx.x;
    if (t >= 2 * O * Kpad) return;
    int c  = t % Kpad;
    int o2 = t / Kpad;
    float v = 0.f;
    if (c < C) {
        if (o2 < O) v = w[(size_t)o2 * (2 * C) + c];
        else {
            int o = o2 - O;
            v = w[(size_t)o * (2 * C) + C + c] - w[(size_t)o * (2 * C) + c];
        }
    }
    Wp[t] = v;
}

// ---- Y[b] = W (M x K4) * X[b] (K4 x N), 16(m) x 64(n) tile per wave
// grid: (N/64, M/16, B), block: 32
__global__ void k_gemm_wmma(const float* __restrict__ W, int ldw,
                            const float* __restrict__ X, int ch_off, int row_stride,
                            int batch_stride, int K4,
                            float* __restrict__ Y, int y_batch_stride) {
    int n0 = blockIdx.x * 64;
    int m0 = blockIdx.y * 16;
    int b  = blockIdx.z;
    int lane = threadIdx.x;
    int r  = lane & 15;
    int hi = lane >> 4;
    const float* Xb = X + (size_t)b * batch_stride + (size_t)ch_off * row_stride;
    v8f acc[4] = {};
    for (int kc = 0; kc < K4; kc += 4) {
        int k0 = kc + 2 * hi;
        const float* r0 = Xb + (size_t)k0 * row_stride;
        const float* r1 = r0 + row_stride;
        v2f a;
        a.x = W[(size_t)(m0 + r) * ldw + k0];
        a.y = W[(size_t)(m0 + r) * ldw + k0 + 1];
#pragma unroll
        for (int j = 0; j < 4; ++j) {
            v2f bb;
            bb.x = r0[n0 + 16 * j + r];
            bb.y = r1[n0 + 16 * j + r];
            acc[j] = __builtin_amdgcn_wmma_f32_16x16x4_f32(false, a, false, bb,
                                                           (short)0, acc[j], false, false);
        }
    }
#pragma unroll
    for (int j = 0; j < 4; ++j) {
#pragma unroll
        for (int v = 0; v < 8; ++v) {
            int m = v + 8 * hi;
            Y[(size_t)b * y_batch_stride + (size_t)(m0 + m) * NPTS + n0 + 16 * j + r] =
                acc[j][v];
        }
    }
}

// ---- out[b, xc_off+o, n] = max_k lrelu( scale[o]*(A[o,idx[n,k]] + Bv[o,n] + bias) + shift[o] )
__global__ void k_gather_max(const float* __restrict__ Yp, int y_batch_stride, int O,
                             const int* __restrict__ idx, const float* __restrict__ bias,
                             const float* __restrict__ g, const float* __restrict__ beta,
                             float* __restrict__ XC, int xc_off) {
    int t = blockIdx.x * blockDim.x + threadIdx.x;
    if (t >= BATCH * O * NPTS) return;
    int n = t % NPTS;
    int o = (t / NPTS) % O;
    int b = t / (NPTS * O);
    const float* A  = Yp + (size_t)b * y_batch_stride + (size_t)o * NPTS;
    const float* Bv = Yp + (size_t)b * y_batch_stride + (size_t)(O + o) * NPTS;
    float scale = g[o] * rsqrtf(1.f + 1e-5f);
    float shift = beta[o];
    float bb = Bv[n] + (bias ? bias[o] : 0.f);
    const int* id = idx + ((size_t)b * NPTS + n) * KNN;
    float best = NEG_INF;
#pragma unroll
    for (int k = 0; k < KNN; ++k) {
        float v = A[id[k]] + bb;
        best = fmaxf(best, lrelu(v * scale + shift));
    }
    XC[((size_t)b * 512 + xc_off + o) * NPTS + n] = best;
}

// ---- gpool[b,o] = max_n E[b,o,n]; one wave per (b,o)
__global__ void k_maxpool(const float* __restrict__ E, float* __restrict__ gpool) {
    int wave = (blockIdx.x * blockDim.x + threadIdx.x) >> 5;
    int lane = threadIdx.x & 31;
    if (wave >= BATCH * 1024) return;
    const float* row = E + (size_t)wave * NPTS;
    float m = NEG_INF;
    for (int j = lane; j < NPTS; j += 32) m = fmaxf(m, row[j]);
#pragma unroll
    for (int off = 16; off > 0; off >>= 1) m = fmaxf(m, __shfl_xor(m, off, 32));
    if (lane == 0) gpool[wave] = m;
}

// ---- head linear: out = act( in @ w^T (+bias) ), act = BN+lrelu if g != null
__global__ void k_linear(const float* __restrict__ in, int Kdim,
                         const float* __restrict__ w, const float* __restrict__ bias,
                         const float* __restrict__ g, const float* __restrict__ beta,
                         int O, float* __restrict__ out) {
    int t = blockIdx.x * blockDim.x + threadIdx.x;
    if (t >= BATCH * O) return;
    int o = t % O;
    int b = t / O;
    const float* row = in + (size_t)b * Kdim;
    const float* wr  = w + (size_t)o * Kdim;
    float s = 0.f;
    for (int c = 0; c < Kdim; ++c) s += row[c] * wr[c];
    if (bias) s += bias[o];
    if (g) s = lrelu(s * (g[o] * rsqrtf(1.f + 1e-5f)) + beta[o]);
    out[t] = s;
}

extern "C" void kernel_launch(void* const* d_in, const int* in_sizes, int n_in,
                              void* d_out, int out_size, void* d_ws, size_t ws_size,
                              hipStream_t stream) {
    (void)in_sizes; (void)n_in; (void)out_size; (void)ws_size;
    const float* x       = (const float*)d_in[0];
    const float* conv1_w = (const float*)d_in[1];
    const float* conv1_b = (const float*)d_in[2];
    const float* bn1_g   = (const float*)d_in[3];
    const float* bn1_b   = (const float*)d_in[4];
    const float* conv2_w = (const float*)d_in[5];
    const float* bn2_g   = (const float*)d_in[6];
    const float* bn2_b   = (const float*)d_in[7];
    const float* conv3_w = (const float*)d_in[8];
    const float* bn3_g   = (const float*)d_in[9];
    const float* bn3_b   = (const float*)d_in[10];
    const float* conv4_w = (const float*)d_in[11];
    const float* bn4_g   = (const float*)d_in[12];
    const float* bn4_b   = (const float*)d_in[13];
    const float* conv5_w = (const float*)d_in[14];
    const float* lin1_w  = (const float*)d_in[15];
    const float* bn6_g   = (const float*)d_in[16];
    const float* bn6_b   = (const float*)d_in[17];
    const float* lin2_w  = (const float*)d_in[18];
    const float* lin2_b  = (const float*)d_in[19];
    const float* bn7_g   = (const float*)d_in[20];
    const float* bn7_b   = (const float*)d_in[21];
    const float* lin3_w  = (const float*)d_in[22];
    const float* lin3_b  = (const float*)d_in[23];

    // workspace layout (floats); total ~= 67.6 MB
    float* ws = (float*)d_ws;
    size_t off = 0;
    float* P    = ws + off; off += (size_t)BATCH * 4 * NPTS;
    float* XX   = ws + off; off += (size_t)BATCH * NPTS;
    float* WP   = ws + off; off += (size_t)512 * 128;
    float* YP   = ws + off; off += (size_t)BATCH * 512 * NPTS;
    float* XC   = ws + off; off += (size_t)BATCH * 512 * NPTS;
    float* DIST = ws + off; off += (size_t)BATCH * NPTS * NPTS;   // reused as E
    float* GP   = ws + off; off += (size_t)BATCH * 1024;
    float* Y1   = ws + off; off += (size_t)BATCH * 512;
    float* Y2   = ws + off; off += (size_t)BATCH * 256;
    int*   IDX  = (int*)(ws + off);

    k_pad_transpose<<<(BATCH * 4 * NPTS + 255) / 256, 256, 0, stream>>>(x, P);

    struct Layer {
        const float* Xsrc; int ch_off, bs, C, O;
        const float *w, *bias, *g, *beta; int xc_off;
    } L[4] = {
        { P,  0,   4 * NPTS,   3,   64,  conv1_w, conv1_b, bn1_g, bn1_b, 0   },
        { XC, 0,   512 * NPTS, 64,  64,  conv2_w, nullptr, bn2_g, bn2_b, 64  },
        { XC, 64,  512 * NPTS, 64,  128, conv3_w, nullptr, bn3_g, bn3_b, 128 },
        { XC, 128, 512 * NPTS, 128, 256, conv4_w, nullptr, bn4_g, bn4_b, 256 },
    };

    for (int li = 0; li < 4; ++li) {
        int C = L[li].C, O = L[li].O;
        int Kpad = (C + 3) & ~3;
        k_sqnorm<<<(BATCH * NPTS + 255) / 256, 256, 0, stream>>>(
            L[li].Xsrc, L[li].ch_off, NPTS, L[li].bs, C, XX);
        k_dist_wmma<<<dim3(NPTS / 64, NPTS / 16, BATCH), 32, 0, stream>>>(
            L[li].Xsrc, L[li].ch_off, NPTS, L[li].bs, Kpad, XX, DIST);
        k_topk<<<(BATCH * NPTS + 127) / 128, 128, 0, stream>>>(DIST, IDX);
        k_prep_w<<<(2 * O * Kpad + 255) / 256, 256, 0, stream>>>(L[li].w, O, C, Kpad, WP);
        k_gemm_wmma<<<dim3(NPTS / 64, (2 * O) / 16, BATCH), 32, 0, stream>>>(
            WP, Kpad, L[li].Xsrc, L[li].ch_off, NPTS, L[li].bs, Kpad, YP, 2 * O * NPTS);
        k_gather_max<<<(BATCH * O * NPTS + 255) / 256, 256, 0, stream>>>(
            YP, 2 * O * NPTS, O, IDX, L[li].bias, L[li].g, L[li].beta, XC, L[li].xc_off);
    }

    // conv5 (1024x512 @ 512x1024 per batch) -> E (reuse DIST), then max over N
    k_gemm_wmma<<<dim3(NPTS / 64, 1024 / 16, BATCH), 32, 0, stream>>>(
        conv5_w, 512, XC, 0, NPTS, 512 * NPTS, 512, DIST, 1024 * NPTS);
    k_maxpool<<<(BATCH * 1024 * 32) / 256, 256, 0, stream>>>(DIST, GP);

    // MLP head
    k_linear<<<(BATCH * 512 + 255) / 256, 256, 0, stream>>>(
        GP, 1024, lin1_w, nullptr, bn6_g, bn6_b, 512, Y1);
    k_linear<<<(BATCH * 256 + 255) / 256, 256, 0, stream>>>(
        Y1, 512, lin2_w, lin2_b, bn7_g, bn7_b, 256, Y2);
    k_linear<<<(BATCH * 40 + 255) / 256, 256, 0, stream>>>(
        Y2, 256, lin3_w, lin3_b, nullptr, nullptr, 40, (float*)d_out);
}